// DilatedAttention_48610439856407
// MI455X (gfx1250) — compile-verified
//
#include <hip/hip_runtime.h>

typedef float v2f __attribute__((ext_vector_type(2)));
typedef float v8f __attribute__((ext_vector_type(8)));

#define N_SEQ 8192
#define HEADS 16
#define DIM   64
#define SEG   2048
#define ROWSTRIDE 3072   // 3 * HEADS * DIM floats per position

__global__ void zero_colsum_kernel(float* cs) {
    int i = blockIdx.x * blockDim.x + threadIdx.x;
    if (i < 4 * HEADS * DIM) cs[i] = 0.0f;
}

// One wave handles 4 consecutive j values => 16 rows (4 j x 4 segments), d = 64.
// Block = 8 waves = 32 j values. Grid = b * h * (2048/32) = 4096 blocks.
__global__ __launch_bounds__(256)
void dilated_attn_main(const float* __restrict__ qkv,
                       float* __restrict__ out,
                       float* __restrict__ colsum) {
    const int lane = threadIdx.x & 31;
    const int wave = threadIdx.x >> 5;
    const int jblk = blockIdx.x & 63;
    const int h    = (blockIdx.x >> 6) & 15;
    const int b    = blockIdx.x >> 10;
    const int j_base = jblk * 32 + wave * 4;

    __shared__ float Stile[8][16][16];
    __shared__ float Ptile[8][16][16];
    __shared__ float bsum[DIM];

    if (threadIdx.x < DIM) bsum[threadIdx.x] = 0.0f;

    const int half = lane >> 4;      // 0 or 1 (lane half)
    const int r    = lane & 15;      // A-row / B-col index
    const int dsub = 2 * half;       // K sub-offset per WMMA f32 16x16x4 layout

    // row r <-> (j_local = r>>2, seg m = r&3); position p = j_base + jl + 2048*m
    const int jl_r = r >> 2, m_r = r & 3;
    const long long p_r = (long long)j_base + jl_r + (long long)SEG * m_r;
    const long long rowbase = ((long long)b * N_SEQ + p_r) * ROWSTRIDE + h * DIM;
    const float* qrow = qkv + rowbase;                 // c = 0 (q)
    const float* krow = qkv + rowbase + HEADS * DIM;   // c = 1 (k)
    const float* vplane = qkv + (long long)b * N_SEQ * ROWSTRIDE + 2 * HEADS * DIM + h * DIM;

    // prefetch this row's V data (used in phase 3)
    __builtin_prefetch(vplane + p_r * ROWSTRIDE, 0, 0);
    __builtin_prefetch(vplane + p_r * ROWSTRIDE + 32, 0, 0);

    // ---- Phase 1: S(16x16) = Q(16x64) * K^T, 16 chained V_WMMA_F32_16X16X4_F32 ----
    v8f s = {};
#pragma unroll
    for (int c = 0; c < 16; ++c) {
        const int dA = 4 * c + dsub;                   // K = g + 2*half within chunk
        v2f a  = *(const v2f*)(qrow + dA);             // A[M=r, dA], A[M=r, dA+1]
        v2f bb = *(const v2f*)(krow + dA);             // B[K, N=r] with same d offsets
        s = __builtin_amdgcn_wmma_f32_16x16x4_f32(false, a, false, bb,
                                                  (short)0, s, false, false);
    }

    // C layout: lane, vgpr g -> M = g + 8*half, N = r
#pragma unroll
    for (int g = 0; g < 8; ++g)
        Stile[wave][g + 8 * half][r] = s[g];
    __syncthreads();

    // ---- Phase 2: 3-branch softmax combine -> block-diagonal weights P ----
    if (lane < 16) {
        const int M  = lane;
        const int mj = M >> 2, mm = M & 3;
        const int j  = j_base + mj;
        const float scale = 0.125f;                    // 1/sqrt(64)
        float sc[4];
#pragma unroll
        for (int i2 = 0; i2 < 4; ++i2) sc[i2] = Stile[wave][M][4 * mj + i2] * scale;

        // branch r=1: softmax over 4 segments
        float mx = fmaxf(fmaxf(sc[0], sc[1]), fmaxf(sc[2], sc[3]));
        float e0 = __expf(sc[0] - mx), e1 = __expf(sc[1] - mx);
        float e2 = __expf(sc[2] - mx), e3 = __expf(sc[3] - mx);
        float inv = 1.0f / (e0 + e1 + e2 + e3);
        float p4[4] = { e0 * inv, e1 * inv, e2 * inv, e3 * inv };

        // branch r=2: if j even, 2-key softmax over pair {mm&1, (mm&1)+2}
        if ((j & 1) == 0) {
            const int a0 = mm & 1, a1 = a0 + 2;
            float m2 = fmaxf(sc[a0], sc[a1]);
            float ea = __expf(sc[a0] - m2), eb = __expf(sc[a1] - m2);
            float inv2 = 1.0f / (ea + eb);
            p4[a0] += ea * inv2;
            p4[a1] += eb * inv2;
        }
        // branch r=4: if j % 4 == 0, identity (softmax over one key)
        if ((j & 3) == 0) p4[mm] += 1.0f;

#pragma unroll
        for (int n = 0; n < 16; ++n) Ptile[wave][M][n] = 0.0f;
#pragma unroll
        for (int i2 = 0; i2 < 4; ++i2) Ptile[wave][M][4 * mj + i2] = p4[i2];
    }
    __syncthreads();

    // ---- Phase 3: O(16x64) = P(16x16) * V(16x64); 4 d-chunks x 4 K-chunks ----
#pragma unroll
    for (int dc = 0; dc < 4; ++dc) {
        v8f o = {};
        const int d = 16 * dc + r;                     // this lane's output column
#pragma unroll
        for (int kc = 0; kc < 4; ++kc) {
            v2f a;                                     // A[M=r, K=4kc+dsub(+1)]
            a.x = Ptile[wave][r][4 * kc + dsub];
            a.y = Ptile[wave][r][4 * kc + dsub + 1];
            // B[K, N=r] = V[row 4kc+K, d]; row (jl=kc, m=K) -> p = j_base + kc + 2048*K
            const long long pA = (long long)j_base + kc + (long long)SEG * dsub;
            const long long pB = pA + SEG;
            v2f bb;
            bb.x = vplane[pA * ROWSTRIDE + d];
            bb.y = vplane[pB * ROWSTRIDE + d];
            o = __builtin_amdgcn_wmma_f32_16x16x4_f32(false, a, false, bb,
                                                      (short)0, o, false, false);
        }
        // store + column-sum partial. C layout: g -> M = g + 8*half
        float part = 0.0f;
#pragma unroll
        for (int g = 0; g < 8; ++g) {
            const int M  = g + 8 * half;
            const int mj = M >> 2, mm = M & 3;
            const long long p = (long long)j_base + mj + (long long)SEG * mm;
            out[(((long long)b * N_SEQ + p) * HEADS + h) * DIM + d] = o[g];
            part += o[g];
        }
        part += __shfl_xor(part, 16, 32);              // combine lane halves (same d)
        if (half == 0) atomicAdd(&bsum[d], part);
    }
    __syncthreads();

    if (threadIdx.x < DIM)
        atomicAdd(&colsum[(b * HEADS + h) * DIM + threadIdx.x], bsum[threadIdx.x]);
}

__global__ void normalize_out(float* __restrict__ out,
                              const float* __restrict__ colsum,
                              long long total) {
    long long i = (long long)blockIdx.x * blockDim.x + threadIdx.x;
    if (i < total) {
        // flat idx: b*2^23 + p*1024 + h*64 + d  ->  colsum idx (b<<10)|(h*64+d)
        const float cs = colsum[((i >> 23) << 10) | (i & 1023)];
        out[i] = out[i] / cs;
    }
}

extern "C" void kernel_launch(void* const* d_in, const int* in_sizes, int n_in,
                              void* d_out, int out_size, void* d_ws, size_t ws_size,
                              hipStream_t stream) {
    const float* qkv = (const float*)d_in[0];
    float* out = (float*)d_out;
    float* colsum = (float*)d_ws;   // 4*16*64 floats = 16 KB

    zero_colsum_kernel<<<16, 256, 0, stream>>>(colsum);
    dilated_attn_main<<<4096, 256, 0, stream>>>(qkv, out, colsum);
    const long long total = (long long)4 * N_SEQ * HEADS * DIM;   // 33,554,432
    normalize_out<<<(int)((total + 255) / 256), 256, 0, stream>>>(out, colsum, total);
}